// GAT_9758165696975
// MI455X (gfx1250) — compile-verified
//
#include <hip/hip_runtime.h>
#include <hip/hip_bf16.h>
#include <math.h>

#define N_NODES     50000
#define N_EDGES     800000
#define IN_FEATS    256
#define HIDDEN      32
#define NHEADS      4
#define NUM_CLASSES 40
#define NEG_SLOPE   0.2f

#define KB       16      // K-depth of one LDS stage (4 WMMA steps)
#define ASTRIDE  20      // padded LDS row stride for the A tile (16 + 4)
#define NMAX     160     // max Nout across layers

typedef __attribute__((ext_vector_type(2))) float v2f;
typedef __attribute__((ext_vector_type(8))) float v8f;

// ---------------------------------------------------------------------------
// Y[M,Nout] = X[M,K] @ W[K,Nout], fp32 WMMA 16x16x4, LDS-staged.
// Grid: M/16 blocks. Block: (Nout/16) waves; wave w computes the 16x16 tile
// at (blockIdx.x, w). Per K-stage: B = 16 consecutive rows of W -> flat
// contiguous copy (b128 coalesced); A = 16x16 block shared by all waves.
// A layout (16x4 f32): lanes 0-15 -> M=lane, VGPRs = K 0/1; lanes 16-31 -> K 2/3.
// C/D layout: VGPR v, lanes 0-15 -> M=v, lanes 16-31 -> M=v+8; N = lane%16.
// Requires: M%16==0, Nout%16==0, K%KB==0, blockDim*8 == 16*Nout (true for
// Nout=128 -> 256 thr and Nout=160 -> 320 thr).
// ---------------------------------------------------------------------------
__global__ void __launch_bounds__(320)
gemm_wmma_f32(const float* __restrict__ X, const float* __restrict__ W,
              float* __restrict__ Y, int K, int Nout) {
  __shared__ float As[16 * ASTRIDE];   // 16 x KB, padded stride
  __shared__ float Bs[KB * NMAX];      // KB x Nout, stride = Nout

  const int t      = threadIdx.x;
  const int lane   = t & 31;
  const int wave   = t >> 5;
  const int nthr   = blockDim.x;
  const int mt     = blockIdx.x;
  const int l16    = lane & 15;
  const int khalf  = (lane >> 4) << 1;   // 0 or 2
  const int ntbase = wave << 4;

  const float* Xblk = X + (mt * 16) * K;   // 16 rows, stride K (int math only)
  v8f acc = (v8f)0.0f;

  for (int k = 0; k < K; k += KB) {
    const float* Wstage = W + k * Nout;    // 16 consecutive rows = flat chunk

    // ---- global -> LDS: B tile, 2 x b128 per thread, fully coalesced ----
    {
      const int i0 = t << 2;               // float index
      *(float4*)&Bs[i0]            = *(const float4*)&Wstage[i0];
      *(float4*)&Bs[i0 + nthr * 4] = *(const float4*)&Wstage[i0 + nthr * 4];
    }
    // ---- global -> LDS: A tile (16x16), 64 threads x b128 (wave-uniform) ----
    if (t < 64) {
      const int r = t >> 2, c = (t & 3) << 2;
      *(float4*)&As[r * ASTRIDE + c] = *(const float4*)&Xblk[r * K + k + c];
    }
    if (k + KB < K)                        // prefetch next B stage (gfx1250)
      __builtin_prefetch(&Wstage[KB * Nout + (t << 2)], 0, 1);
    __syncthreads();

    // ---- 4 WMMA steps out of LDS ----
#pragma unroll
    for (int kk = 0; kk < KB; kk += 4) {
      v2f a = *(const v2f*)&As[l16 * ASTRIDE + kk + khalf];
      v2f b;
      b.x = Bs[(kk + khalf    ) * Nout + ntbase + l16];
      b.y = Bs[(kk + khalf + 1) * Nout + ntbase + l16];
      acc = __builtin_amdgcn_wmma_f32_16x16x4_f32(false, a, false, b,
                                                  (short)0, acc, false, false);
    }
    __syncthreads();
  }

  float* yrow = Y + (mt * 16 + ((lane >> 4) << 3)) * Nout + ntbase + l16;
#pragma unroll
  for (int v = 0; v < 8; ++v)
    yrow[v * Nout] = acc[v];
}

// ---------------------------------------------------------------------------
// el[n,h] = sum_d feat[n,h,d]*al[h,d]; er likewise. One thread per (n,h).
// ---------------------------------------------------------------------------
__global__ void attn_scores(const float* __restrict__ feat,
                            const float* __restrict__ al,
                            const float* __restrict__ ar,
                            float* __restrict__ el, float* __restrict__ er,
                            int N, int H, int D) {
  int i = blockIdx.x * blockDim.x + threadIdx.x;   // i = n*H + h
  if (i >= N * H) return;
  int h = i % H;
  const float* f = feat + (size_t)i * (size_t)D;
  const float* a = al + h * D;
  const float* b = ar + h * D;
  float sl = 0.f, sr = 0.f;
  for (int d = 0; d < D; ++d) {
    float v = f[d];
    sl = fmaf(v, a[d], sl);
    sr = fmaf(v, b[d], sr);
  }
  el[i] = sl;
  er[i] = sr;
}

// ---------------------------------------------------------------------------
// row_ptr[n] = lower_bound(dst, n) for sorted dst. One thread per n (0..N).
// ---------------------------------------------------------------------------
__global__ void build_row_ptr(const int* __restrict__ dst,
                              int* __restrict__ row_ptr, int N, int E) {
  int n = blockIdx.x * blockDim.x + threadIdx.x;
  if (n > N) return;
  int lo = 0, hi = E;
  while (lo < hi) {
    int mid = (lo + hi) >> 1;
    if (dst[mid] < n) lo = mid + 1; else hi = mid;
  }
  row_ptr[n] = lo;
}

// ---------------------------------------------------------------------------
// Per-destination-node edge softmax + weighted aggregation (deterministic,
// dst is sorted -> CSR segments). One block per node, blockDim = H*D.
// mode 0: write elu(acc) to out[n*H*D + t]
// mode 2: mean over heads, write out[n*D + d] (final logits)
// ---------------------------------------------------------------------------
__global__ void gat_aggregate(const float* __restrict__ feat,   // [N,H*D]
                              const float* __restrict__ el,     // [N,H]
                              const float* __restrict__ er,     // [N,H]
                              const int* __restrict__ src,      // [E]
                              const int* __restrict__ row_ptr,  // [N+1]
                              float* __restrict__ out,
                              int H, int D, int mode) {
  __shared__ float red[160];
  __shared__ float mh[4];
  __shared__ float sh[4];

  const int n  = blockIdx.x;
  const int t  = threadIdx.x;           // 0 .. H*D-1
  const int HD = blockDim.x;
  const int e0 = row_ptr[n];
  const int e1 = row_ptr[n + 1];

  // ---- pass 1: per-head max of leaky_relu(el[src]+er[n]) -------------------
  const int hh     = t % H;             // head for the edge-parallel passes
  const int estep  = HD / H;
  const float ernh = er[(size_t)n * H + hh];

  float lmax = -INFINITY;
  for (int e = e0 + t / H; e < e1; e += estep) {
    float x = el[(size_t)src[e] * H + hh] + ernh;
    x = (x > 0.f) ? x : x * NEG_SLOPE;
    lmax = fmaxf(lmax, x);
  }
  red[t] = lmax;
  __syncthreads();
  if (t < H) {
    float m = -INFINITY;
    for (int i = t; i < HD; i += H) m = fmaxf(m, red[i]);
    mh[t] = m;
  }
  __syncthreads();

  // ---- pass 2: per-head sum of exp(x - m) ---------------------------------
  const float mhh = mh[hh];
  float lsum = 0.f;
  for (int e = e0 + t / H; e < e1; e += estep) {
    float x = el[(size_t)src[e] * H + hh] + ernh;
    x = (x > 0.f) ? x : x * NEG_SLOPE;
    lsum += expf(x - mhh);
  }
  __syncthreads();               // red[] about to be reused
  red[t] = lsum;
  __syncthreads();
  if (t < H) {
    float s = 0.f;
    for (int i = t; i < HD; i += H) s += red[i];
    sh[t] = s;
  }
  __syncthreads();

  // ---- pass 3: acc = sum_e alpha_e * feat[src_e, h, d], thread t = (h,d) ---
  const int   h    = t / D;
  const float ern  = er[(size_t)n * H + h];
  const float m    = mh[h];
  const float dnm  = sh[h];
  const float invd = (dnm > 0.f) ? (1.0f / dnm) : 0.f;

  float acc = 0.f;
  for (int e = e0; e < e1; ++e) {
    int s = src[e];
    float x = el[(size_t)s * H + h] + ern;
    x = (x > 0.f) ? x : x * NEG_SLOPE;
    float alpha = expf(x - m) * invd;
    acc = fmaf(feat[(size_t)s * HD + t], alpha, acc);
  }

  if (mode == 2) {               // final layer: mean over heads -> logits
    __syncthreads();
    red[t] = acc;
    __syncthreads();
    if (t < D) {
      float s = 0.f;
      for (int h2 = 0; h2 < H; ++h2) s += red[h2 * D + t];
      out[(size_t)n * D + t] = s * (1.0f / (float)H);
    }
  } else {                       // ELU activation
    out[(size_t)n * HD + t] = (acc > 0.f) ? acc : expm1f(acc);
  }
}

// ---------------------------------------------------------------------------
extern "C" void kernel_launch(void* const* d_in, const int* in_sizes, int n_in,
                              void* d_out, int out_size, void* d_ws, size_t ws_size,
                              hipStream_t stream) {
  (void)in_sizes; (void)n_in; (void)out_size; (void)ws_size;

  const float* h   = (const float*)d_in[0];
  const int*   src = (const int*)d_in[1];
  const int*   dst = (const int*)d_in[2];
  const float* W0  = (const float*)d_in[3];
  const float* al0 = (const float*)d_in[4];
  const float* ar0 = (const float*)d_in[5];
  const float* W1  = (const float*)d_in[6];
  const float* al1 = (const float*)d_in[7];
  const float* ar1 = (const float*)d_in[8];
  const float* W2  = (const float*)d_in[9];
  const float* al2 = (const float*)d_in[10];
  const float* ar2 = (const float*)d_in[11];
  float* out = (float*)d_out;

  // workspace layout (bytes, 256-aligned):
  //   feat : 50000*160 f32 = 32,000,000
  //   xbuf : 50000*128 f32 = 25,600,000
  //   el   : 50000*4   f32 =    800,000
  //   er   : 50000*4   f32 =    800,000
  //   rowp : 50001     i32 =    200,004      total ~59.4 MB
  char* ws = (char*)d_ws;
  float* feat = (float*)(ws + 0);
  float* xbuf = (float*)(ws + 32000000);
  float* el   = (float*)(ws + 57600000);
  float* er   = (float*)(ws + 58400000);
  int*   rowp = (int*)  (ws + 59200000);

  build_row_ptr<<<(N_NODES + 1 + 255) / 256, 256, 0, stream>>>(dst, rowp,
                                                               N_NODES, N_EDGES);

  const int MT = N_NODES / 16;   // 3125 M-tiles

  // -------- layer 0: in 256 -> 4 heads x 32 --------
  {
    const int K = IN_FEATS, Nf = NHEADS * HIDDEN;           // 256 -> 128
    gemm_wmma_f32<<<MT, (Nf / 16) * 32, 0, stream>>>(h, W0, feat, K, Nf);
    attn_scores<<<(N_NODES * NHEADS + 255) / 256, 256, 0, stream>>>(
        feat, al0, ar0, el, er, N_NODES, NHEADS, HIDDEN);
    gat_aggregate<<<N_NODES, NHEADS * HIDDEN, 0, stream>>>(
        feat, el, er, src, rowp, xbuf, NHEADS, HIDDEN, 0);
  }

  // -------- layer 1: 128 -> 4 heads x 32 --------
  {
    const int K = NHEADS * HIDDEN, Nf = NHEADS * HIDDEN;    // 128 -> 128
    gemm_wmma_f32<<<MT, (Nf / 16) * 32, 0, stream>>>(xbuf, W1, feat, K, Nf);
    attn_scores<<<(N_NODES * NHEADS + 255) / 256, 256, 0, stream>>>(
        feat, al1, ar1, el, er, N_NODES, NHEADS, HIDDEN);
    gat_aggregate<<<N_NODES, NHEADS * HIDDEN, 0, stream>>>(
        feat, el, er, src, rowp, xbuf, NHEADS, HIDDEN, 0);
  }

  // -------- layer 2: 128 -> 4 heads x 40, then mean over heads --------
  {
    const int K = NHEADS * HIDDEN, Nf = NHEADS * NUM_CLASSES; // 128 -> 160
    gemm_wmma_f32<<<MT, (Nf / 16) * 32, 0, stream>>>(xbuf, W2, feat, K, Nf);
    attn_scores<<<(N_NODES * NHEADS + 255) / 256, 256, 0, stream>>>(
        feat, al2, ar2, el, er, N_NODES, NHEADS, NUM_CLASSES);
    gat_aggregate<<<N_NODES, NHEADS * NUM_CLASSES, 0, stream>>>(
        feat, el, er, src, rowp, out, NHEADS, NUM_CLASSES, 2);
  }
}